// TemporalEncoder_13125420056981
// MI455X (gfx1250) — compile-verified
//
#include <hip/hip_runtime.h>
#include <math.h>

typedef __attribute__((ext_vector_type(2))) float v2f;
typedef __attribute__((ext_vector_type(4))) float v4f;
typedef __attribute__((ext_vector_type(8))) float v8f;

#define DIM 8192

// ---------------------------------------------------------------------------
// Kernel 1: single workgroup. Max-reduce x[0..8191], then spikes[i] =
// (int(floor(x[i]/xmax * 255)) == t) ? 1.0f : 0.0f   (tau == 1.0)
// spikes lives in d_ws (32 KB) and stays hot in L2 for kernel 2.
// ---------------------------------------------------------------------------
__global__ void __launch_bounds__(256)
spike_mask_kernel(const float* __restrict__ x,
                  const int* __restrict__ t_ptr,
                  float* __restrict__ spikes) {
  __shared__ float red[256];
  const int tid = threadIdx.x;

  float m = 0.0f;  // x is uniform(0,1), all non-negative
  for (int i = tid; i < DIM; i += 256) m = fmaxf(m, x[i]);
  red[tid] = m;
  __syncthreads();
  for (int s = 128; s > 0; s >>= 1) {
    if (tid < s) red[tid] = fmaxf(red[tid], red[tid + s]);
    __syncthreads();
  }
  const float xmax = red[0];
  const int t = *t_ptr;

  for (int i = tid; i < DIM; i += 256) {
    // match reference op order: floor((x / max) * 255 / tau)
    const float q = floorf((x[i] / xmax) * 255.0f);
    spikes[i] = ((int)q == t) ? 1.0f : 0.0f;
  }
}

// ---------------------------------------------------------------------------
// Kernel 2: OUT = W x diag(spikes), one wave (32 lanes) per 16x16 tile.
// 512x512 tiles, 8 waves per 256-thread block -> 32768 blocks, exact cover
// (EXEC all-1s, as WMMA requires).
//
// All-zero spike tiles (~255/256 of them with t=128): skip the weight read
// entirely, stream zeros with b128 nontemporal stores -> traffic approaches
// the 256 MB write-only floor (~11 us at 23.3 TB/s).
//
// Nonzero tiles: exact f32 path via V_WMMA_F32_16X16X4_F32, K=16 split into
// 4 chunks of K=4.
//   A (16x4 f32, 2 VGPR): lane L holds row M=L&15, K = 4q + {0,1} (L<16)
//                                                or K = 4q + {2,3} (L>=16)
//   B (4x16 f32, 2 VGPR): VGPR0 rows kk={0,2}, VGPR1 rows kk={1,3};
//                         diag chunk: B[kk][n] = (n == 4q+kk) ? spikes[n] : 0
//   C/D (16x16 f32, 8 VGPR): VGPR j -> M = j + (L>=16 ? 8 : 0), N = L&15
// ---------------------------------------------------------------------------
__global__ void __launch_bounds__(256)
diag_scale_wmma_kernel(const float* __restrict__ w,
                       const float* __restrict__ spikes,
                       float* __restrict__ out) {
  const int wave  = (int)((blockIdx.x * blockDim.x + threadIdx.x) >> 5);
  const int lane  = (int)(threadIdx.x & 31);
  const int ct    = wave & 511;   // column-tile index (fast -> adjacent waves
  const int rt    = wave >> 9;    // share rows, contiguous HBM segments)
  const int cbase = ct << 4;
  const int rbase = rt << 4;
  const int nn    = lane & 15;
  const bool up   = lane >= 16;

  const float s = spikes[cbase + nn];  // lane's column's spike (L2-resident)

  // wave-uniform predicate -> no divergence, EXEC stays all-1s for WMMA
  if (__ballot(s != 0.0f) != 0ull) {
    v8f c = {};
#pragma unroll
    for (int q = 0; q < 4; ++q) {
      const int kk0 = 4 * q + (up ? 2 : 0);
      // A chunk: row (rbase+nn), cols (cbase+kk0, cbase+kk0+1); 8B aligned
      const float* ap = w + (size_t)(rbase + nn) * DIM + cbase + kk0;
      v2f a;
      a.x = __builtin_nontemporal_load(ap);
      a.y = __builtin_nontemporal_load(ap + 1);
      // B chunk: diagonal slice in registers
      v2f b;
      b.x = (nn == kk0)     ? s : 0.0f;
      b.y = (nn == kk0 + 1) ? s : 0.0f;
      c = __builtin_amdgcn_wmma_f32_16x16x4_f32(
          /*neg_a=*/false, a, /*neg_b=*/false, b,
          /*c_mod=*/(short)0, c, /*reuse_a=*/false, /*reuse_b=*/false);
    }
#pragma unroll
    for (int j = 0; j < 8; ++j) {
      const int row = rbase + j + (up ? 8 : 0);
      __builtin_nontemporal_store(c[j], out + (size_t)row * DIM + cbase + nn);
    }
  } else {
    // zero tile: 2x b128 nontemporal stores per lane (256 floats / wave)
    const int r  = rbase + (lane >> 1);
    const int cc = cbase + (lane & 1) * 8;
    float* p = out + (size_t)r * DIM + cc;
    v4f z = {0.0f, 0.0f, 0.0f, 0.0f};
    __builtin_nontemporal_store(z, (v4f*)p);
    __builtin_nontemporal_store(z, (v4f*)(p + 4));
  }
}

// ---------------------------------------------------------------------------
extern "C" void kernel_launch(void* const* d_in, const int* in_sizes, int n_in,
                              void* d_out, int out_size, void* d_ws, size_t ws_size,
                              hipStream_t stream) {
  const float* x      = (const float*)d_in[0];   // [1, 8192] f32
  const float* weight = (const float*)d_in[1];   // [8192, 8192] f32
  const int*   t      = (const int*)d_in[2];     // scalar int
  float* out    = (float*)d_out;                 // [8192, 8192] f32
  float* spikes = (float*)d_ws;                  // 8192 f32 scratch (32 KB)

  spike_mask_kernel<<<1, 256, 0, stream>>>(x, t, spikes);

  const int tiles  = (DIM / 16) * (DIM / 16);    // 262144 waves
  const int blocks = tiles / (256 / 32);         // 8 waves per block -> 32768
  diag_scale_wmma_kernel<<<blocks, 256, 0, stream>>>(weight, spikes, out);
}